// Model_52381421142380
// MI455X (gfx1250) — compile-verified
//
#include <hip/hip_runtime.h>
#include <math.h>

// ---------------------------------------------------------------------------
// Model dims
// ---------------------------------------------------------------------------
#define BS    256
#define TLEN  3
#define NN    90
#define SS    65
#define CHEV  45
#define TF    10
#define FEAT  (88*43)        // 3784
#define FLAT  (TLEN*FEAT)    // 11352
#define H1    1024
#define H2    256
#define NCLS  2

// padded dims (WMMA-friendly, all even)
#define NP    96             // 90 -> 96  (M pad)
#define SP    68             // 65 -> 68  (K pad, mult of 4)
#define CP    48             // 45 -> 48  (N pad, mult of 16)
#define KADJ  92             // 90 -> 92  (K pad for adj GEMM)

typedef __attribute__((ext_vector_type(2))) float v2f;
typedef __attribute__((ext_vector_type(8))) float v8f;

// ---------------------------------------------------------------------------
// WMMA f32 16x16x4 helpers.
// A frag: lanes 0-15: M=row0+lane, K=k0..k0+1 ; lanes 16-31: M=row0+lane-16, K=k0+2..k0+3
// C frag: VGPR r -> M = row0 + r (+8 for lanes>=16), N = col0 + (lane&15)
// All operands padded -> every load/store unconditional (EXEC stays all-ones).
// ---------------------------------------------------------------------------
static __device__ inline v8f wmma4(v2f a, v2f b, v8f c) {
  return __builtin_amdgcn_wmma_f32_16x16x4_f32(
      false, a, false, b, (short)0, c, false, false);
}

// lda and k0 even -> 8-byte aligned pair load
static __device__ inline v2f load_a2(const float* __restrict__ A, int lda,
                                     int row0, int k0, int lane) {
  int m  = row0 + (lane & 15);
  int kb = k0 + ((lane >> 4) << 1);
  return *(const v2f*)(A + (size_t)m * lda + kb);
}

static __device__ inline v2f load_b2(const float* __restrict__ B, int ldb,
                                     int k0, int col0, int lane) {
  int n  = col0 + (lane & 15);
  int kb = k0 + ((lane >> 4) << 1);
  v2f b;
  b.x = B[(size_t)kb * ldb + n];
  b.y = B[(size_t)(kb + 1) * ldb + n];
  return b;
}

// ---------------------------------------------------------------------------
// K0a/K0b: stage inputs into zero-padded workspace copies
// ---------------------------------------------------------------------------
__global__ __launch_bounds__(256)
void k_pad_fdata(const float* __restrict__ src, float* __restrict__ dst) {
  int idx = blockIdx.x * 256 + threadIdx.x;          // over 768*96*68
  int s = idx % SP, rem = idx / SP;
  int n = rem % NP, bt = rem / NP;
  dst[idx] = (n < NN && s < SS) ? src[((size_t)bt * NN + n) * SS + s] : 0.0f;
}

__global__ __launch_bounds__(256)
void k_pad_theta(const float* __restrict__ src, float* __restrict__ dst) {
  int idx = blockIdx.x * 256 + threadIdx.x;          // over 3*68*48
  if (idx >= 3 * SP * CP) return;
  int o = idx % CP, rem = idx / CP;
  int s = rem % SP, k = rem / SP;
  dst[idx] = (s < SS && o < CHEV) ? src[((size_t)k * SS + s) * CHEV + o] : 0.0f;
}

// ---------------------------------------------------------------------------
// K1: uncertainty head: cs_sum[b] = sum_t sigmoid( <fdata[b,t,:,:], un_w> + un_b )
// ---------------------------------------------------------------------------
__global__ __launch_bounds__(256)
void k_uncert(const float* __restrict__ fdata, const float* __restrict__ un_w,
              const float* __restrict__ un_b, float* __restrict__ cs_sum) {
  __shared__ float red[256];
  int b = blockIdx.x, tid = threadIdx.x;
  float total = 0.0f;
  for (int t = 0; t < TLEN; ++t) {
    const float* X = fdata + ((size_t)b * TLEN + t) * (NN * SS);
    float s = 0.0f;
    for (int i = tid; i < NN * SS; i += 256) s += X[i] * un_w[i];
    red[tid] = s; __syncthreads();
    for (int st = 128; st > 0; st >>= 1) {
      if (tid < st) red[tid] += red[tid + st];
      __syncthreads();
    }
    if (tid == 0) total += 1.0f / (1.0f + __expf(-(red[0] + un_b[0])));
    __syncthreads();
  }
  if (tid == 0) cs_sum[b] = total;
}

// ---------------------------------------------------------------------------
// K2: per (t,b): pearson -> W -> L -> power iteration -> T1,T2 (padded 96x92)
// ---------------------------------------------------------------------------
__global__ __launch_bounds__(256)
void k_adj(const float* __restrict__ fdata, float* __restrict__ adjT1,
           float* __restrict__ adjT2) {
  __shared__ float xs[NN][SS + 1];
  __shared__ float Wm[NN][NN + 1];
  __shared__ float mu[NN], inr[NN], vv[NN], wv[NN];
  __shared__ float lam_s;

  int t = blockIdx.x / BS, b = blockIdx.x % BS;
  const float* X = fdata + ((size_t)b * TLEN + t) * (NN * SS);
  int tid = threadIdx.x;

  for (int i = tid; i < NN * SS; i += 256) xs[i / SS][i % SS] = X[i];
  __syncthreads();

  if (tid < NN) {
    float s = 0.0f;
    for (int j = 0; j < SS; ++j) s += xs[tid][j];
    float m = s / (float)SS;
    float nrm = 0.0f;
    for (int j = 0; j < SS; ++j) { float d = xs[tid][j] - m; nrm += d * d; }
    mu[tid] = m;
    inr[tid] = 1.0f / (sqrtf(nrm) + 1e-6f);
  }
  __syncthreads();
  for (int i = tid; i < NN * SS; i += 256) {
    int r = i / SS, c = i % SS;
    xs[r][c] = (xs[r][c] - mu[r]) * inr[r];
  }
  __syncthreads();

  for (int idx = tid; idx < NN * NN; idx += 256) {   // W = (corr+1)/2
    int i = idx / NN, j = idx % NN;
    float s = 0.0f;
    for (int c = 0; c < SS; ++c) s += xs[i][c] * xs[j][c];
    Wm[i][j] = (s + 1.0f) * 0.5f;
  }
  __syncthreads();

  if (tid < NN) {                                    // degree
    float s = 0.0f;
    for (int j = 0; j < NN; ++j) s += Wm[tid][j];
    mu[tid] = s;
  }
  __syncthreads();
  for (int idx = tid; idx < NN * NN; idx += 256) {   // L = D - W
    int i = idx / NN, j = idx % NN;
    Wm[i][j] = ((i == j) ? mu[i] : 0.0f) - Wm[i][j];
  }
  if (tid < NN) vv[tid] = 1.0f;
  __syncthreads();

  for (int it = 0; it < 32; ++it) {                  // power iteration
    if (tid < NN) {
      float s = 0.0f;
      for (int j = 0; j < NN; ++j) s += Wm[tid][j] * vv[j];
      wv[tid] = s;
    }
    __syncthreads();
    if (tid == 0) {
      float s = 0.0f;
      for (int i = 0; i < NN; ++i) s += wv[i] * wv[i];
      lam_s = sqrtf(s);
    }
    __syncthreads();
    float invl = 1.0f / fmaxf(lam_s, 1e-20f);
    if (tid < NN) vv[tid] = wv[tid] * invl;
    __syncthreads();
  }

  float lam = fmaxf(lam_s, 1e-20f);
  float* T1 = adjT1 + (size_t)blockIdx.x * (NP * KADJ);
  float* T2 = adjT2 + (size_t)blockIdx.x * (NP * KADJ);
  for (int idx = tid; idx < NP * KADJ; idx += 256) {
    int i = idx / KADJ, j = idx % KADJ;
    if (i < NN && j < NN) {
      float eye = (i == j) ? 1.0f : 0.0f;
      float t1 = 2.0f * Wm[i][j] / lam - eye;
      T1[idx] = t1;
      T2[idx] = 2.0f * t1 * t1 - eye;
    } else {
      T1[idx] = 0.0f;
      T2[idx] = 0.0f;
    }
  }
}

// ---------------------------------------------------------------------------
// K3: F[bt][k] (96x48) = fdp[bt] (96x68) @ thp[k] (68x48)        (WMMA)
//     one wave = 16x48 strip (A reused across 3 accumulators)
// ---------------------------------------------------------------------------
__global__ __launch_bounds__(256)
void k_ftheta(const float* __restrict__ fdp, const float* __restrict__ thp,
              float* __restrict__ F) {
  int wid  = (blockIdx.x * 256 + threadIdx.x) >> 5;
  int lane = threadIdx.x & 31;
  int batch = wid / 6;                 // bt*3 + k
  if (batch >= BS * TLEN * 3) return;
  int tm = wid % 6;
  int k  = batch % 3;
  int bt = batch / 3;

  const float* A = fdp + (size_t)bt * (NP * SP);
  const float* B = thp + (size_t)k * (SP * CP);
  float* C = F + (size_t)batch * (NP * CP);

  v8f a0 = {}, a1 = {}, a2 = {};
  for (int k0 = 0; k0 < SP; k0 += 4) {
    v2f a  = load_a2(A, SP, tm * 16, k0, lane);
    v2f b0 = load_b2(B, CP, k0, 0,  lane);
    v2f b1 = load_b2(B, CP, k0, 16, lane);
    v2f b2 = load_b2(B, CP, k0, 32, lane);
    a0 = wmma4(a, b0, a0);
    a1 = wmma4(a, b1, a1);
    a2 = wmma4(a, b2, a2);
  }
  int n = lane & 15;
  int mbase = tm * 16 + ((lane >> 4) << 3);
  #pragma unroll
  for (int r = 0; r < 8; ++r) {
    float* row = C + (size_t)(mbase + r) * CP;
    row[n]      = a0[r];
    row[16 + n] = a1[r];
    row[32 + n] = a2[r];
  }
}

// ---------------------------------------------------------------------------
// K4: h[ta,b,tp] = relu( T1@F1 + T2@F2 + F0 ), all padded 96x48    (WMMA)
// ---------------------------------------------------------------------------
__global__ __launch_bounds__(256)
void k_h(const float* __restrict__ adjT1, const float* __restrict__ adjT2,
         const float* __restrict__ F, float* __restrict__ H) {
  int wid  = (blockIdx.x * 256 + threadIdx.x) >> 5;
  int lane = threadIdx.x & 31;
  int batch = wid / 6;                 // ta*768 + b*3 + tp
  if (batch >= TLEN * BS * TLEN) return;
  int tm = wid % 6;
  int ta = batch / (BS * TLEN);
  int bt = batch % (BS * TLEN);        // b*3 + tp
  int b  = bt / TLEN;

  v8f a0 = {}, a1 = {}, a2 = {};
  #pragma unroll
  for (int kk = 1; kk <= 2; ++kk) {
    const float* A = ((kk == 1) ? adjT1 : adjT2) +
                     (size_t)(ta * BS + b) * (NP * KADJ);
    const float* B = F + ((size_t)bt * 3 + kk) * (NP * CP);
    for (int k0 = 0; k0 < KADJ; k0 += 4) {
      v2f a  = load_a2(A, KADJ, tm * 16, k0, lane);
      v2f b0 = load_b2(B, CP, k0, 0,  lane);
      v2f b1 = load_b2(B, CP, k0, 16, lane);
      v2f b2 = load_b2(B, CP, k0, 32, lane);
      a0 = wmma4(a, b0, a0);
      a1 = wmma4(a, b1, a1);
      a2 = wmma4(a, b2, a2);
    }
  }

  const float* F0 = F + ((size_t)bt * 3 + 0) * (NP * CP);
  float* C = H + (size_t)batch * (NP * CP);
  int n = lane & 15;
  int mbase = tm * 16 + ((lane >> 4) << 3);
  #pragma unroll
  for (int r = 0; r < 8; ++r) {
    size_t base = (size_t)(mbase + r) * CP;
    C[base + n]      = fmaxf(a0[r] + F0[base + n],      0.0f);
    C[base + 16 + n] = fmaxf(a1[r] + F0[base + 16 + n], 0.0f);
    C[base + 32 + n] = fmaxf(a2[r] + F0[base + 32 + n], 0.0f);
  }
}

// ---------------------------------------------------------------------------
// K5: fused tconv(3x3,C=3->10)+relu+1x1(10->1) + cs_sum scale -> x flat
//     H padded layout: [ta*BS+b][tp][96][48]
// ---------------------------------------------------------------------------
__global__ __launch_bounds__(256)
void k_conv(const float* __restrict__ H, const float* __restrict__ tw,
            const float* __restrict__ tb, const float* __restrict__ ow,
            const float* __restrict__ ob, const float* __restrict__ cs_sum,
            float* __restrict__ x) {
  __shared__ float hs[TLEN][NN][CHEV];
  __shared__ float tws[TF * TLEN * 9];
  __shared__ float tbs[TF], ows[TF];
  __shared__ float obs, csb;

  int gid = blockIdx.x;                          // ta*BS + b
  int ta = gid / BS, b = gid % BS;
  const float* Hp = H + (size_t)gid * (TLEN * NP * CP);
  int tid = threadIdx.x;

  for (int i = tid; i < TLEN * NN * CHEV; i += 256) {
    int q = i % CHEV, rem = i / CHEV;
    int r = rem % NN, c = rem / NN;
    hs[c][r][q] = Hp[((size_t)c * NP + r) * CP + q];
  }
  if (tid < TF * TLEN * 9) tws[tid] = tw[tid];
  if (tid < TF) { tbs[tid] = tb[tid]; ows[tid] = ow[tid]; }
  if (tid == 0) { obs = ob[0]; csb = cs_sum[b]; }
  __syncthreads();

  for (int p = tid; p < FEAT; p += 256) {
    int pi = p / 43, qi = p % 43;
    float acc = obs;
    #pragma unroll
    for (int tf = 0; tf < TF; ++tf) {
      float s = tbs[tf];
      #pragma unroll
      for (int c = 0; c < TLEN; ++c)
        #pragma unroll
        for (int di = 0; di < 3; ++di)
          #pragma unroll
          for (int dj = 0; dj < 3; ++dj)
            s += tws[((tf * TLEN + c) * 3 + di) * 3 + dj] * hs[c][pi + di][qi + dj];
      acc += ows[tf] * fmaxf(s, 0.0f);
    }
    x[(size_t)b * FLAT + (size_t)ta * FEAT + p] = acc * csb;
  }
}

// ---------------------------------------------------------------------------
// K6: y1 = BN( x (256x11352) @ l1_w (11352x1024) + b )   (WMMA, 16x64/wave)
// ---------------------------------------------------------------------------
__global__ __launch_bounds__(256)
void k_l1(const float* __restrict__ x, const float* __restrict__ w,
          const float* __restrict__ bias, const float* __restrict__ g,
          const float* __restrict__ beta, float* __restrict__ y) {
  int wid  = (blockIdx.x * 256 + threadIdx.x) >> 5;
  int lane = threadIdx.x & 31;
  if (wid >= (BS / 16) * (H1 / 64)) return;
  int tm   = wid / (H1 / 64);
  int col0 = (wid % (H1 / 64)) * 64;

  v8f a0 = {}, a1 = {}, a2 = {}, a3 = {};
  for (int k0 = 0; k0 < FLAT; k0 += 4) {
    v2f a  = load_a2(x, FLAT, tm * 16, k0, lane);
    v2f b0 = load_b2(w, H1, k0, col0,      lane);
    v2f b1 = load_b2(w, H1, k0, col0 + 16, lane);
    v2f b2 = load_b2(w, H1, k0, col0 + 32, lane);
    v2f b3 = load_b2(w, H1, k0, col0 + 48, lane);
    a0 = wmma4(a, b0, a0);
    a1 = wmma4(a, b1, a1);
    a2 = wmma4(a, b2, a2);
    a3 = wmma4(a, b3, a3);
  }

  const float bnsc = rsqrtf(1.0f + 1e-5f);
  int mbase = tm * 16 + ((lane >> 4) << 3);
  v8f accs[4] = {a0, a1, a2, a3};
  #pragma unroll
  for (int j = 0; j < 4; ++j) {
    int n = col0 + j * 16 + (lane & 15);
    float sc = g[n] * bnsc, bt = beta[n], bi = bias[n];
    #pragma unroll
    for (int r = 0; r < 8; ++r)
      y[(size_t)(mbase + r) * H1 + n] = (accs[j][r] + bi) * sc + bt;
  }
}

// ---------------------------------------------------------------------------
// K7: y2 = BN( y1 (256x1024) @ l2_w (1024x256) + b )     (WMMA, 16x64/wave)
// ---------------------------------------------------------------------------
__global__ __launch_bounds__(256)
void k_l2(const float* __restrict__ x, const float* __restrict__ w,
          const float* __restrict__ bias, const float* __restrict__ g,
          const float* __restrict__ beta, float* __restrict__ y) {
  int wid  = (blockIdx.x * 256 + threadIdx.x) >> 5;
  int lane = threadIdx.x & 31;
  if (wid >= (BS / 16) * (H2 / 64)) return;
  int tm   = wid / (H2 / 64);
  int col0 = (wid % (H2 / 64)) * 64;

  v8f a0 = {}, a1 = {}, a2 = {}, a3 = {};
  for (int k0 = 0; k0 < H1; k0 += 4) {
    v2f a  = load_a2(x, H1, tm * 16, k0, lane);
    v2f b0 = load_b2(w, H2, k0, col0,      lane);
    v2f b1 = load_b2(w, H2, k0, col0 + 16, lane);
    v2f b2 = load_b2(w, H2, k0, col0 + 32, lane);
    v2f b3 = load_b2(w, H2, k0, col0 + 48, lane);
    a0 = wmma4(a, b0, a0);
    a1 = wmma4(a, b1, a1);
    a2 = wmma4(a, b2, a2);
    a3 = wmma4(a, b3, a3);
  }

  const float bnsc = rsqrtf(1.0f + 1e-5f);
  int mbase = tm * 16 + ((lane >> 4) << 3);
  v8f accs[4] = {a0, a1, a2, a3};
  #pragma unroll
  for (int j = 0; j < 4; ++j) {
    int n = col0 + j * 16 + (lane & 15);
    float sc = g[n] * bnsc, bt = beta[n], bi = bias[n];
    #pragma unroll
    for (int r = 0; r < 8; ++r)
      y[(size_t)(mbase + r) * H2 + n] = (accs[j][r] + bi) * sc + bt;
  }
}

// ---------------------------------------------------------------------------
// K8: logits = y2 @ l3_w + b ; softmax
// ---------------------------------------------------------------------------
__global__ __launch_bounds__(256)
void k_head(const float* __restrict__ y2, const float* __restrict__ w,
            const float* __restrict__ b, float* __restrict__ out) {
  int i = blockIdx.x * 256 + threadIdx.x;
  if (i >= BS) return;
  float l0 = b[0], l1 = b[1];
  for (int k = 0; k < H2; ++k) {
    float v = y2[(size_t)i * H2 + k];
    l0 += v * w[k * NCLS + 0];
    l1 += v * w[k * NCLS + 1];
  }
  float m = fmaxf(l0, l1);
  float e0 = __expf(l0 - m), e1 = __expf(l1 - m);
  float inv = 1.0f / (e0 + e1);
  out[(size_t)i * NCLS + 0] = e0 * inv;
  out[(size_t)i * NCLS + 1] = e1 * inv;
}

// ---------------------------------------------------------------------------
// Launch
// ---------------------------------------------------------------------------
extern "C" void kernel_launch(void* const* d_in, const int* in_sizes, int n_in,
                              void* d_out, int out_size, void* d_ws, size_t ws_size,
                              hipStream_t stream) {
  const float* fdata   = (const float*)d_in[0];
  const float* un_w    = (const float*)d_in[1];
  const float* un_b    = (const float*)d_in[2];
  const float* theta   = (const float*)d_in[3];
  const float* tconv_w = (const float*)d_in[4];
  const float* tconv_b = (const float*)d_in[5];
  const float* oconv_w = (const float*)d_in[6];
  const float* oconv_b = (const float*)d_in[7];
  const float* l1_w    = (const float*)d_in[8];
  const float* l1_b    = (const float*)d_in[9];
  const float* bn1_g   = (const float*)d_in[10];
  const float* bn1_b   = (const float*)d_in[11];
  const float* l2_w    = (const float*)d_in[12];
  const float* l2_b    = (const float*)d_in[13];
  const float* bn2_g   = (const float*)d_in[14];
  const float* bn2_b   = (const float*)d_in[15];
  const float* l3_w    = (const float*)d_in[16];
  const float* l3_b    = (const float*)d_in[17];
  float* out = (float*)d_out;

  float* ws = (float*)d_ws;
  size_t off = 0;
  float* cs_sum = ws + off; off += 256;
  float* fdp    = ws + off; off += (size_t)BS * TLEN * NP * SP;          // 5.0 M
  float* thp    = ws + off; off += (size_t)3 * SP * CP;
  float* adjT1  = ws + off; off += (size_t)TLEN * BS * NP * KADJ;        // 6.8 M
  float* adjT2  = ws + off; off += (size_t)TLEN * BS * NP * KADJ;        // 6.8 M
  float* F      = ws + off; off += (size_t)BS * TLEN * 3 * NP * CP;      // 10.6 M
  float* H      = ws + off; off += (size_t)TLEN * BS * TLEN * NP * CP;   // 10.6 M
  float* xbuf   = ws + off; off += (size_t)BS * FLAT;                    // 2.9 M
  float* y1     = ws + off; off += (size_t)BS * H1;
  float* y2     = ws + off; off += (size_t)BS * H2;

  {
    int tot = BS * TLEN * NP * SP;                      // 5013504 = 256*19584
    k_pad_fdata<<<tot / 256, 256, 0, stream>>>(fdata, fdp);
    k_pad_theta<<<(3 * SP * CP + 255) / 256, 256, 0, stream>>>(theta, thp);
  }

  k_uncert<<<BS, 256, 0, stream>>>(fdata, un_w, un_b, cs_sum);
  k_adj<<<TLEN * BS, 256, 0, stream>>>(fdata, adjT1, adjT2);

  {  // 2304 batches x 6 strip-tiles = 13824 waves -> 1728 blocks
    int blocks = (BS * TLEN * 3 * 6) / 8;
    k_ftheta<<<blocks, 256, 0, stream>>>(fdp, thp, F);
    k_h<<<blocks, 256, 0, stream>>>(adjT1, adjT2, F, H);
  }

  k_conv<<<TLEN * BS, 256, 0, stream>>>(H, tconv_w, tconv_b, oconv_w, oconv_b,
                                        cs_sum, xbuf);

  k_l1<<<((BS / 16) * (H1 / 64)) / 8, 256, 0, stream>>>(xbuf, l1_w, l1_b,
                                                        bn1_g, bn1_b, y1);
  k_l2<<<((BS / 16) * (H2 / 64)) / 8, 256, 0, stream>>>(y1, l2_w, l2_b,
                                                        bn2_g, bn2_b, y2);
  k_head<<<1, 256, 0, stream>>>(y2, l3_w, l3_b, out);
  (void)in_sizes; (void)n_in; (void)out_size; (void)ws_size;
}